// ClusterLookup_v2_17042430230775
// MI455X (gfx1250) — compile-verified
//
#include <hip/hip_runtime.h>
#include <hip/hip_bf16.h>

typedef __attribute__((ext_vector_type(2))) float v2f;
typedef __attribute__((ext_vector_type(8))) float v8f;

#define N_ROWS 131072
#define DIM    512
#define KCLUST 64
#define ROWS_PER_BLOCK 128
#define CHUNK  32            // D-chunk staged in LDS per outer iteration
#define CPITCH 160           // floats per k-pair row in LDS (32-bank offset between rows)
#define EPSN   1e-12f

// ---------------------------------------------------------------------------
// Kernel 1: L2-normalize clusters [64,512], store transposed + k-PAIR-INTERLEAVED:
//   ct2[(d>>1)*128 + k*2 + (d&1)]   (d = dim 0..511, k = cluster 0..63)
// so a WMMA B-fragment (k, k+1 for one n) is a contiguous 8-byte load.
// ---------------------------------------------------------------------------
__global__ void k_norm_clusters(const float* __restrict__ clusters,
                                float* __restrict__ ct2) {
    __shared__ float sred[8];
    const int k    = blockIdx.x;          // cluster index, 64 blocks
    const int tid  = threadIdx.x;         // 256 threads
    const int lane = tid & 31;
    const int wid  = tid >> 5;

    float v0 = clusters[k * DIM + tid];
    float v1 = clusters[k * DIM + tid + 256];
    float ss = v0 * v0 + v1 * v1;
    #pragma unroll
    for (int o = 16; o > 0; o >>= 1) ss += __shfl_xor(ss, o, 32);
    if (lane == 0) sred[wid] = ss;
    __syncthreads();
    float tot = 0.f;
    #pragma unroll
    for (int i = 0; i < 8; ++i) tot += sred[i];
    const float inv = 1.0f / fmaxf(sqrtf(tot), EPSN);

    const int d0 = tid, d1 = tid + 256;
    ct2[(d0 >> 1) * 128 + k * 2 + (d0 & 1)] = v0 * inv;
    ct2[(d1 >> 1) * 128 + k * 2 + (d1 & 1)] = v1 * inv;
}

// ---------------------------------------------------------------------------
// Kernel 2: fused GEMM (f32 WMMA) + row-norm scaling + softmax + loss partials
// Staging issues all 6 global b128 loads before any LDS store (6x MLP/wave,
// temps die before the barrier -> no cross-region register pressure/spills).
// Block: 256 threads (8 waves), 128 rows of x. Grid: 1024.
// ---------------------------------------------------------------------------
__global__ void __launch_bounds__(256)
k_cluster_probs(const float* __restrict__ x,
                const float* __restrict__ ct2,     // pair-interleaved normalized^T
                const float* __restrict__ alphap,
                float* __restrict__ out_probs,     // d_out + 1
                float* __restrict__ loss_part) {   // [gridDim.x]
    __shared__ float xs[ROWS_PER_BLOCK * 36];      // x chunk, pitch 36 (conflict-free frags)
    __shared__ float cs[(CHUNK / 2) * CPITCH];     // 16 k-pair rows x 128 (+pad to 160)
    __shared__ float tile[ROWS_PER_BLOCK * 65];    // inner/probs tile, pitch 65
    __shared__ float lred[8];

    const int tid   = threadIdx.x;
    const int lane  = tid & 31;
    const int wave  = tid >> 5;          // 0..7
    const int laneN = lane & 15;         // N / M index within fragment
    const int laneH = lane >> 4;         // half: selects K pair
    const int rowbase = blockIdx.x * ROWS_PER_BLOCK;
    const float alpha = alphap[0];

    // Per-thread staging coordinates (constant across chunks)
    const int xr0 = tid >> 3;                      // x: row for idx=tid
    const int xc0 = (tid & 7) << 2;                //    col
    const int pr0 = tid >> 5;                      // ct: pair-row for idx=tid
    const int q0  = (tid & 31) << 2;               //     offset

    v8f acc[4];
    #pragma unroll
    for (int t = 0; t < 4; ++t) acc[t] = (v8f){0.f,0.f,0.f,0.f,0.f,0.f,0.f,0.f};
    float ss = 0.f;                       // partial sum-of-squares for row laneN

    for (int kb = 0; kb < DIM; kb += CHUNK) {
        // ---- stage: issue ALL global loads first (6 in flight per wave) ----
        float4 a0 = *(const float4*)&x[(size_t)(rowbase + xr0)       * DIM + kb + xc0];
        float4 a1 = *(const float4*)&x[(size_t)(rowbase + xr0 +  32) * DIM + kb + xc0];
        float4 a2 = *(const float4*)&x[(size_t)(rowbase + xr0 +  64) * DIM + kb + xc0];
        float4 a3 = *(const float4*)&x[(size_t)(rowbase + xr0 +  96) * DIM + kb + xc0];
        float4 c0 = *(const float4*)&ct2[(size_t)((kb >> 1) + pr0)     * 128 + q0];
        float4 c1 = *(const float4*)&ct2[(size_t)((kb >> 1) + pr0 + 8) * 128 + q0];

        // ---- then commit to LDS (compiler staggers the loadcnt waits) ----
        *(float4*)&xs[(xr0      ) * 36 + xc0] = a0;
        *(float4*)&xs[(xr0 +  32) * 36 + xc0] = a1;
        *(float4*)&xs[(xr0 +  64) * 36 + xc0] = a2;
        *(float4*)&xs[(xr0 +  96) * 36 + xc0] = a3;
        *(float4*)&cs[(pr0    ) * CPITCH + q0] = c0;
        *(float4*)&cs[(pr0 + 8) * CPITCH + q0] = c1;
        __syncthreads();

        // ---- WMMA over this chunk: 8 k-steps x 4 column tiles ----
        #pragma unroll
        for (int kk = 0; kk < CHUNK; kk += 4) {
            // A fragment (16x4 f32): lane holds row laneN, k = kk + 2*laneH + {0,1}
            v2f a = *(const v2f*)&xs[(wave * 16 + laneN) * 36 + kk + 2 * laneH];
            ss += a.x * a.x + a.y * a.y;
            const int prow = (kk >> 1) + laneH;        // k-pair row in cs
            #pragma unroll
            for (int t = 0; t < 4; ++t) {
                // B fragment: contiguous pair (k, k+1) for column n -> single b64
                v2f b = *(const v2f*)&cs[prow * CPITCH + (t * 16 + laneN) * 2];
                acc[t] = __builtin_amdgcn_wmma_f32_16x16x4_f32(
                    false, a, false, b, (short)0, acc[t], false, false);
            }
        }
        __syncthreads();
    }

    // ---- row 1/||x|| : combine the two lane-halves ----
    ss += __shfl_xor(ss, 16, 32);
    const float rinv = 1.0f / fmaxf(sqrtf(ss), EPSN);   // for row laneN

    // ---- scale + spill 16x64 tile to LDS (C/D layout: M = g + 8*laneH, N = laneN) ----
    #pragma unroll
    for (int t = 0; t < 4; ++t) {
        #pragma unroll
        for (int g = 0; g < 8; ++g) {
            float invm = __shfl(rinv, g + 8 * laneH, 32);
            tile[(wave * 16 + g + 8 * laneH) * 65 + t * 16 + laneN] = acc[t][g] * invm;
        }
    }
    __syncthreads();

    // ---- softmax over K=64 per row: 2 lanes per row, 32 cols each ----
    const int srow  = tid >> 1;           // 0..127  (wave w handles rows 16w..16w+15)
    const int shalf = tid & 1;
    float* rp = &tile[srow * 65 + shalf * 32];

    float m = -3.402823466e+38f;
    #pragma unroll
    for (int i = 0; i < 32; ++i) m = fmaxf(m, alpha * rp[i]);
    m = fmaxf(m, __shfl_xor(m, 1, 32));

    float se = 0.f, sev = 0.f;
    #pragma unroll
    for (int i = 0; i < 32; ++i) {
        float v = rp[i];
        float e = __expf(alpha * v - m);
        se  += e;
        sev += e * v;
    }
    float totv = se + __shfl_xor(se, 1, 32);
    float itot = 1.0f / totv;
    #pragma unroll
    for (int i = 0; i < 32; ++i) {
        float v = rp[i];
        rp[i] = __expf(alpha * v - m) * itot;
    }
    float lpart = sev * itot;             // this half-row's sum(p * inner)
    __syncthreads();

    // ---- coalesced store of probs tile: 128x64 ----
    #pragma unroll
    for (int p = 0; p < 32; ++p) {
        int idx = tid + p * 256;          // 0..8191
        int r   = idx >> 6;
        int c   = idx & 63;
        out_probs[(size_t)(rowbase + r) * KCLUST + c] = tile[r * 65 + c];
    }

    // ---- deterministic per-block loss partial ----
    #pragma unroll
    for (int o = 16; o > 0; o >>= 1) lpart += __shfl_xor(lpart, o, 32);
    if (lane == 0) lred[wave] = lpart;
    __syncthreads();
    if (tid == 0) {
        float t2 = 0.f;
        #pragma unroll
        for (int i = 0; i < 8; ++i) t2 += lred[i];
        loss_part[blockIdx.x] = t2;
    }
}

// ---------------------------------------------------------------------------
// Kernel 3: reduce 1024 block partials -> loss = -sum/N (deterministic)
// ---------------------------------------------------------------------------
__global__ void k_finalize(const float* __restrict__ parts, float* __restrict__ out) {
    __shared__ float sred[8];
    const int tid  = threadIdx.x;        // 256
    const int lane = tid & 31;
    const int wid  = tid >> 5;
    float s = parts[tid] + parts[tid + 256] + parts[tid + 512] + parts[tid + 768];
    #pragma unroll
    for (int o = 16; o > 0; o >>= 1) s += __shfl_xor(s, o, 32);
    if (lane == 0) sred[wid] = s;
    __syncthreads();
    if (tid == 0) {
        float tot = 0.f;
        #pragma unroll
        for (int i = 0; i < 8; ++i) tot += sred[i];
        out[0] = -tot / (float)N_ROWS;
    }
}

// ---------------------------------------------------------------------------
extern "C" void kernel_launch(void* const* d_in, const int* in_sizes, int n_in,
                              void* d_out, int out_size, void* d_ws, size_t ws_size,
                              hipStream_t stream) {
    (void)in_sizes; (void)n_in; (void)out_size; (void)ws_size;
    const float* x        = (const float*)d_in[0];   // [131072, 512]
    const float* clusters = (const float*)d_in[1];   // [64, 512]
    const float* alpha    = (const float*)d_in[2];   // scalar
    float* out = (float*)d_out;                      // [1 + 131072*64]

    float* ct2   = (float*)d_ws;                     // [256*128] pair-interleaved ĉᵀ
    float* parts = ct2 + DIM * KCLUST;               // [1024] loss partials

    k_norm_clusters<<<KCLUST, 256, 0, stream>>>(clusters, ct2);
    k_cluster_probs<<<N_ROWS / ROWS_PER_BLOCK, 256, 0, stream>>>(
        x, ct2, alpha, out + 1, parts);
    k_finalize<<<1, 256, 0, stream>>>(parts, out);
}